// TopoLossMSE3D_2740189134901
// MI455X (gfx1250) — compile-verified
//
#include <hip/hip_runtime.h>

// TopoLossMSE3D: loss = sum( ((pred>.5)!=(tgt>.5)) * (pred - (tgt>.5))^2 )
// Pure bandwidth-bound streaming reduction: ~262 MB in, 4 B out (~11 us floor
// at 23.3 TB/s). Strategy: b128 NT loads -> per-lane accumulate ->
// V_WMMA_F32_16X16X4_F32 wave reduction -> deterministic two-pass block sum.

typedef __attribute__((ext_vector_type(2))) float v2f;
typedef __attribute__((ext_vector_type(4))) float f32x4;
typedef __attribute__((ext_vector_type(8))) float v8f;

// Full-wave (32-lane) sum using the CDNA5 matrix pipe.
// A (16x4 f32, 2 VGPRs): lanes 0-15 hold A[M=lane][K=0]=p, K=1 -> 0;
//                        lanes 16-31 hold A[M=lane-16][K=2]=p, K=3 -> 0.
// B (4x16 f32, 2 VGPRs): all ones.
// D[m][n] = sum_k A[m][k] = p_m + p_{m+16}; D VGPR j: lanes 0-15 -> M=j,
// lanes 16-31 -> M=8+j. Sum of the 8 D VGPRs per lane gives the half-sums;
// one xor-16 shuffle completes the 32-lane total (broadcast to all lanes).
// Requires EXEC all-ones: call only at reconverged points.
__device__ __forceinline__ float wave_reduce_wmma(float p) {
    v2f a; a.x = p;    a.y = 0.0f;
    v2f b; b.x = 1.0f; b.y = 1.0f;
    v8f c = {};
    c = __builtin_amdgcn_wmma_f32_16x16x4_f32(
        /*neg_a=*/false, a, /*neg_b=*/false, b,
        /*c_mod=*/(short)0, c, /*reuse_a=*/false, /*reuse_b=*/false);
    float s = ((c[0] + c[1]) + (c[2] + c[3])) + ((c[4] + c[5]) + (c[6] + c[7]));
    s += __shfl_xor(s, 16, 32);
    return s;
}

__device__ __forceinline__ float elem_contrib(float p, float t) {
    const bool pb = p > 0.5f;
    const bool tb = t > 0.5f;
    const float forced = tb ? 1.0f : 0.0f;
    const float d = p - forced;
    return (pb != tb) ? d * d : 0.0f;
}

__global__ void topo_partial_kernel(const float* __restrict__ pred,
                                    const float* __restrict__ tgt,
                                    float* __restrict__ partial,
                                    long long n) {
    const long long n4     = n >> 2;
    const long long tid    = (long long)blockIdx.x * blockDim.x + threadIdx.x;
    const long long stride = (long long)gridDim.x * blockDim.x;
    const f32x4* p4 = (const f32x4*)pred;
    const f32x4* t4 = (const f32x4*)tgt;

    float acc = 0.0f;
    for (long long i = tid; i < n4; i += stride) {
        const f32x4 p = __builtin_nontemporal_load(&p4[i]);
        const f32x4 t = __builtin_nontemporal_load(&t4[i]);
        acc += elem_contrib(p.x, t.x);
        acc += elem_contrib(p.y, t.y);
        acc += elem_contrib(p.z, t.z);
        acc += elem_contrib(p.w, t.w);
    }
    // Scalar tail (n not multiple of 4): handled by block 0 only.
    if (blockIdx.x == 0) {
        const long long base = n4 << 2;
        if ((long long)threadIdx.x < (n - base)) {
            acc += elem_contrib(pred[base + threadIdx.x], tgt[base + threadIdx.x]);
        }
    }

    // Reconverged here: EXEC all-ones for the WMMA reduction.
    const float wsum = wave_reduce_wmma(acc);

    __shared__ float lds[32];
    const unsigned lane = threadIdx.x & 31u;
    const unsigned wave = threadIdx.x >> 5;
    if (lane == 0) lds[wave] = wsum;
    __syncthreads();
    if (threadIdx.x == 0) {
        const unsigned nw = (blockDim.x + 31u) >> 5;
        float bs = 0.0f;
        for (unsigned w = 0; w < nw; ++w) bs += lds[w];  // fixed order: deterministic
        partial[blockIdx.x] = bs;
    }
}

// One wave, fixed strided order per lane + WMMA wave reduce: deterministic.
__global__ void topo_final_kernel(const float* __restrict__ partial,
                                  float* __restrict__ out, int nb) {
    float acc = 0.0f;
    for (int i = (int)threadIdx.x; i < nb; i += 32) acc += partial[i];
    const float s = wave_reduce_wmma(acc);
    if (threadIdx.x == 0) out[0] = s;
}

extern "C" void kernel_launch(void* const* d_in, const int* in_sizes, int n_in,
                              void* d_out, int out_size, void* d_ws, size_t ws_size,
                              hipStream_t stream) {
    const float* pred = (const float*)d_in[0];
    const float* tgt  = (const float*)d_in[1];
    float* out        = (float*)d_out;
    float* partial    = (float*)d_ws;
    const long long n = (long long)in_sizes[0];

    int nb = 2048;  // 2048 blocks x 8 waves: plenty to saturate 23.3 TB/s
    const long long ws_cap = (long long)(ws_size / sizeof(float));
    if ((long long)nb > ws_cap) nb = (int)(ws_cap > 0 ? ws_cap : 1);

    topo_partial_kernel<<<nb, 256, 0, stream>>>(pred, tgt, partial, n);
    topo_final_kernel<<<1, 32, 0, stream>>>(partial, out, nb);
}